// ExpVariational_44332652429426
// MI455X (gfx1250) — compile-verified
//
#include <hip/hip_runtime.h>

// out_b = x_b @ expm(t_b * K),  K = diag(sigma) + skew-tridiag, 128x128.
// Substepped Taylor ACTION (never materializing expm):
//   h = t/SUBSTEPS;  y <- sum_{k=0..TERMS} v_k,  v_k = (h/k) v_{k-1} K,  SUBSTEPS times.
// K-apply = dense 16x128 x 128x128 GEMM via V_WMMA_F32_16X16X4_F32 (full f32 precision).
// K (transposed) lives in registers per wave; v_k tile double-buffers in LDS.

#define UNITS     128
#define IN_STRIDE 129          // inputs row: 128 state + 1 time
#define TILE      16           // rows per workgroup
#define SUBSTEPS  8            // h*||K|| <= ~2
#define TERMS     12           // tail ~ (h||K||)^13/13! ~ 1e-7
#define YSTRIDE   132          // padded LDS row stride (floats) -> conflict-free A loads

typedef __attribute__((ext_vector_type(2))) float v2f;
typedef __attribute__((ext_vector_type(8))) float v8f;

// Build K transposed (KdT[j*128+i] = K[i][j]) into workspace.
// sigma_i = -softplus(p_i)^2 ; ceta_i = -p[128+i]
// K[i][i]=sigma_i ; K[i][i+1]=ceta_i ; K[i][i-1]=-ceta_{i-1}
__global__ void build_KT(const float* __restrict__ p, float* __restrict__ KdT) {
    int idx = blockIdx.x * blockDim.x + threadIdx.x;
    if (idx >= UNITS * UNITS) return;
    int i = idx >> 7;    // row of K
    int j = idx & 127;   // col of K
    float val = 0.0f;
    if (j == i) {
        float x  = p[i];
        float sp = (x > 20.0f) ? x : log1pf(__expf(x));
        val = -sp * sp;
    } else if (j == i + 1) {
        val = -p[UNITS + i];        // ceta_i
    } else if (j + 1 == i) {
        val = p[UNITS + j];         // -ceta_j
    }
    KdT[j * UNITS + i] = val;
}

__global__ void __launch_bounds__(256)
expmv_taylor_wmma(const float* __restrict__ inp,   // [B,129]
                  const float* __restrict__ KdT,   // [128,128] transposed
                  float* __restrict__ out,         // [B,128]
                  int B)
{
    __shared__ float Ybuf[2][TILE * YSTRIDE];      // double-buffered v_k tile (16x128 f32)

    const int tid   = threadIdx.x;
    const int wave  = tid >> 5;        // 8 wave32s
    const int lane  = tid & 31;
    const int col   = lane & 15;       // N (and A-row M) index within 16
    const int hi    = lane >> 4;       // lane-half: K pair select / row+8
    const int jbase = wave * 16;       // this wave's 16-column block
    const int row0  = blockIdx.x * TILE;

    // ---- Preload all B fragments of K for this wave's column block (loop invariant).
    // B(4x16) layout: lane holds N=lane&15; VGPR0 = K={0|2}, VGPR1 = K={1|3} per half.
    // KdT is column-major K -> the two K values are consecutive -> one b64 load each.
    v2f breg[32];
    #pragma unroll
    for (int c = 0; c < 32; ++c) {
        breg[c] = *reinterpret_cast<const v2f*>(
            &KdT[(jbase + col) * UNITS + 4 * c + 2 * hi]);
    }

    // ---- Per-lane row-aligned data for C/D layout: reg v <-> row (v + 8*hi).
    float tvec[8];
    v8f S = {};   // Taylor-sum block (this wave's 16x16), register-resident
    #pragma unroll
    for (int v = 0; v < 8; ++v) {
        int r  = row0 + v + 8 * hi;
        int rc = (r < B) ? r : (B - 1);
        tvec[v] = inp[rc * IN_STRIDE + UNITS];             // t_b
        S[v]    = inp[rc * IN_STRIDE + jbase + col];       // v_0 block = x block
    }

    // ---- Cooperative fill of Ybuf[0] with the x tile (v_0).
    for (int idx = tid; idx < TILE * UNITS; idx += 256) {
        int r  = idx >> 7, cc = idx & 127;
        int rr = row0 + r;
        int rc = (rr < B) ? rr : (B - 1);
        Ybuf[0][r * YSTRIDE + cc] = inp[rc * IN_STRIDE + cc];
    }
    __syncthreads();

    const int aoff = col * YSTRIDE + 2 * hi;   // base of this lane's A fragments

    int cur = 0;
    for (int step = 0; step < SUBSTEPS; ++step) {
        for (int k = 1; k <= TERMS; ++k) {
            const float* Yc = Ybuf[cur];
            float*       Yn = Ybuf[cur ^ 1];

            // Z = (v_{k-1} * K)[:, jbase:jbase+16]
            // Software-pipelined: 8 A-fragments staged in registers so ~4
            // ds_load_2addr_b64 stay in flight while WMMAs drain older ones.
            v8f z0 = {};
            v8f z1 = {};
            v2f abuf[8];
            #pragma unroll
            for (int c = 0; c < 8; ++c)
                abuf[c] = *reinterpret_cast<const v2f*>(&Yc[aoff + 4 * c]);
            #pragma unroll
            for (int c = 0; c < 32; ++c) {
                v2f a = abuf[c & 7];
                if (c + 8 < 32)
                    abuf[c & 7] = *reinterpret_cast<const v2f*>(&Yc[aoff + 4 * (c + 8)]);
                if (c & 1)
                    z1 = __builtin_amdgcn_wmma_f32_16x16x4_f32(
                            false, a, false, breg[c], (short)0, z1, false, false);
                else
                    z0 = __builtin_amdgcn_wmma_f32_16x16x4_f32(
                            false, a, false, breg[c], (short)0, z0, false, false);
            }

            // v_k = (t_b / (SUBSTEPS*k)) * Z  (per-row scale); accumulate Taylor sum.
            // v_rcp_f32 (1 ulp) on the scalar coefficient -- far below f32 accum noise.
            const float invsk = __builtin_amdgcn_rcpf((float)(SUBSTEPS * k));
            const bool  last  = (k == TERMS);
            #pragma unroll
            for (int v = 0; v < 8; ++v) {
                float val = (z0[v] + z1[v]) * tvec[v] * invsk;
                S[v] += val;
                // Last term: publish S (the substep result) as next substep's v_0,
                // folding the substep-tail writeback into this store (saves a barrier).
                Yn[(v + 8 * hi) * YSTRIDE + jbase + col] = last ? S[v] : val;
            }
            __syncthreads();
            cur ^= 1;
        }
    }

    // ---- Write result (C/D layout back to row-major global).
    #pragma unroll
    for (int v = 0; v < 8; ++v) {
        int r = row0 + v + 8 * hi;
        if (r < B) out[r * UNITS + jbase + col] = S[v];
    }
}

extern "C" void kernel_launch(void* const* d_in, const int* in_sizes, int n_in,
                              void* d_out, int out_size, void* d_ws, size_t ws_size,
                              hipStream_t stream) {
    const float* inp    = (const float*)d_in[0];   // [B, 129] f32
    const float* params = (const float*)d_in[1];   // [255] f32
    float*       out    = (float*)d_out;           // [B, 128] f32
    float*       KdT    = (float*)d_ws;            // 128*128 f32 = 64 KB scratch

    int B = in_sizes[0] / IN_STRIDE;

    build_KT<<<(UNITS * UNITS + 255) / 256, 256, 0, stream>>>(params, KdT);

    int tiles = (B + TILE - 1) / TILE;
    expmv_taylor_wmma<<<tiles, 256, 0, stream>>>(inp, KdT, out, B);
}